// KVCacheAttention_22273700397074
// MI455X (gfx1250) — compile-verified
//
#include <hip/hip_runtime.h>
#include <stdint.h>

typedef float v2f __attribute__((ext_vector_type(2)));
typedef float v8f __attribute__((ext_vector_type(8)));

#define B_   16
#define T_   8
#define C_   2048
#define H_   16
#define DK_  128
#define S_   4096
#define ST_  4104      // S + T
#define N3C_ 6144      // 3*C
#define NEGINF (-3.0e38f)

// LDS layout for attn kernel (floats). K/V staging rows padded 128->132 floats
// so B-fragment reads (stride 132: bank = 4*ln mod 64) are conflict-free.
#define KVROW_   132
#define KVBUF_   (16 * KVROW_)         // 2112 floats per K or V buffer
#define SM_KV_   (8 * 2 * KVBUF_)      // 33792
#define SM_P_    (SM_KV_)              // p tiles: 8 waves x 256
#define SM_CTX_  (SM_P_ + 8 * 256)     // 35840
#define SM_M_    (SM_CTX_ + 2048)      // 37888
#define SM_L_    (SM_M_ + 16)          // 37904
#define SM_TOT_  (SM_L_ + 16)          // 37920 floats = 151680 bytes

__device__ __forceinline__ float rmax16(float v) {
  #pragma unroll
  for (int m = 1; m < 16; m <<= 1) v = fmaxf(v, __shfl_xor(v, m, 16));
  return v;
}
__device__ __forceinline__ float rsum16(float v) {
  #pragma unroll
  for (int m = 1; m < 16; m <<= 1) v += __shfl_xor(v, m, 16);
  return v;
}

// ---------------------------------------------------------------------------
// QKV projection: x[128,2048] @ w_qkv[2048,6144] + b_qkv.
// Scatters q -> qbuf[B,H,T,dk], k_new/v_new -> tail (s = S..S+T-1) of ko/vo.
// ---------------------------------------------------------------------------
__global__ __launch_bounds__(256)
void qkv_gemm_kernel(const float* __restrict__ x, const float* __restrict__ w,
                     const float* __restrict__ bias, float* __restrict__ qbuf,
                     float* __restrict__ ko, float* __restrict__ vo) {
  const int wave = threadIdx.x >> 5;
  const int lane = threadIdx.x & 31;
  const int tile = blockIdx.x * 8 + wave;          // 3072 tiles = 8 mt x 384 nt
  const int nt = tile % 384;
  const int mt = tile / 384;
  const int ln = lane & 15;
  const int koff = (lane >> 4) << 1;
  const int arow = mt * 16 + ln;
  const int bcol = nt * 16 + ln;

  v8f acc = {};
  for (int k = 0; k < C_; k += 4) {
    v2f a = *reinterpret_cast<const v2f*>(x + arow * C_ + k + koff);
    v2f b;
    b.x = w[(k + koff) * N3C_ + bcol];
    b.y = w[(k + koff + 1) * N3C_ + bcol];
    acc = __builtin_amdgcn_wmma_f32_16x16x4_f32(false, a, false, b, (short)0,
                                                acc, false, false);
  }

  const float bv = bias[bcol];
  const int rbase = mt * 16 + ((lane >> 4) << 3);
  #pragma unroll
  for (int r = 0; r < 8; ++r) {
    float val = acc[r] + bv;
    int row = rbase + r;                 // row = b*T + t
    int b_ = row >> 3;
    int t  = row & 7;
    int which = bcol >> 11;              // 0=q, 1=k, 2=v
    int rem = bcol & 2047;
    int h  = rem >> 7;
    int dd = rem & 127;
    if (which == 0) {
      qbuf[((b_ * H_ + h) * T_ + t) * DK_ + dd] = val;
    } else {
      float* dst = (which == 1) ? ko : vo;
      dst[((long long)(b_ * H_ + h) * ST_ + (S_ + t)) * DK_ + dd] = val;
    }
  }
}

// ---------------------------------------------------------------------------
// Fused cache-concat + flash attention. One block per (b,h); 8 waves.
// Past chunks: async-DMA global->LDS, WMMA from LDS, async-DMA LDS->k/v cache
// (the concat write-through). Tail chunk: guarded direct-global path.
// ---------------------------------------------------------------------------
__global__ __launch_bounds__(256)
void attn_kernel(const float* __restrict__ qbuf,
                 const float* __restrict__ pk, const float* __restrict__ pv,
                 float* __restrict__ ko, float* __restrict__ vo,
                 float* __restrict__ ctxbuf) {
  extern __shared__ float smem[];

  const int bh   = blockIdx.x;
  const int wave = threadIdx.x >> 5;
  const int lane = threadIdx.x & 31;
  const int ln   = lane & 15;
  const int hi   = lane >> 4;
  const int koff = hi << 1;

  float* kbuf   = smem + wave * (2 * KVBUF_);
  float* vbuf   = kbuf + KVBUF_;
  float* myp    = smem + SM_P_ + wave * 256;
  float* ctx_sh = smem + SM_CTX_;
  float* m_sh   = smem + SM_M_;
  float* l_sh   = smem + SM_L_;

  // LDS byte offsets for the async DMA engine (low 32 bits of flat address)
  const uint32_t klds = (uint32_t)(uintptr_t)kbuf + (uint32_t)lane * 16u;
  const uint32_t vlds = (uint32_t)(uintptr_t)vbuf + (uint32_t)lane * 16u;

  const float* Q = qbuf + bh * (T_ * DK_);

  // Q as 32 A-fragments (K-steps of 4); rows 8..15 zero-padded.
  v2f qa[32];
  {
    v2f z = {0.f, 0.f};
    #pragma unroll
    for (int i = 0; i < 32; ++i)
      qa[i] = (ln < T_) ? *reinterpret_cast<const v2f*>(Q + ln * DK_ + i * 4 + koff) : z;
  }

  float m[8], l[8];
  v8f acc[8];
  #pragma unroll
  for (int r = 0; r < 8; ++r) { m[r] = NEGINF; l[r] = 0.f; }
  {
    v8f z = {};
    #pragma unroll
    for (int nd = 0; nd < 8; ++nd) acc[nd] = z;
  }

  const float scale = 0.08838834764831845f;     // 1/sqrt(128)
  const int NCH = (ST_ + 15) >> 4;              // 257

  for (int ch = wave; ch < NCH; ch += 8) {
    const int s0 = ch << 4;
    const bool wr = (s0 < S_);                  // uniform per chunk
    const bool sok = (s0 + ln) < ST_;           // only tail chunk partial

    const float* kp = nullptr;
    const float* vp = nullptr;

    if (wr) {
      const uint64_t kg  = (uint64_t)(uintptr_t)(pk + ((long long)bh * S_ + s0) * DK_);
      const uint64_t vg  = (uint64_t)(uintptr_t)(pv + ((long long)bh * S_ + s0) * DK_);

      // previous chunk's LDS->global DMA must finish before reusing buffers
      asm volatile("s_wait_asynccnt 0x0" ::: "memory");

      // stage 16x128 f32 K and V rows: row j (512B) -> padded LDS row (528B)
      #pragma unroll
      for (int j = 0; j < 16; ++j) {
        uint32_t goff = (uint32_t)lane * 16u + (uint32_t)j * 512u;
        asm volatile("global_load_async_to_lds_b128 %0, %1, %2"
                     :: "v"(klds + (uint32_t)j * 528u), "v"(goff), "s"(kg)
                     : "memory");
        asm volatile("global_load_async_to_lds_b128 %0, %1, %2"
                     :: "v"(vlds + (uint32_t)j * 528u), "v"(goff), "s"(vg)
                     : "memory");
      }
      asm volatile("s_wait_asynccnt 0x0" ::: "memory");
    } else {
      kp = ko + ((long long)bh * ST_ + s0) * DK_;   // qkv-written tail
      vp = vo + ((long long)bh * ST_ + s0) * DK_;
    }

    // ---- scores: Q (A) x K^T (B) ----
    v8f sc = {};
    #pragma unroll
    for (int i = 0; i < 32; ++i) {
      v2f b;
      if (wr) {
        b = *reinterpret_cast<const v2f*>(kbuf + ln * KVROW_ + i * 4 + koff);
      } else {
        b = (v2f){0.f, 0.f};
        if (sok) b = *reinterpret_cast<const v2f*>(kp + ln * DK_ + i * 4 + koff);
      }
      sc = __builtin_amdgcn_wmma_f32_16x16x4_f32(false, qa[i], false, b,
                                                 (short)0, sc, false, false);
    }

    if (wr) {
      // cache-concat write-through: coalesced async DMA LDS -> ko/vo
      const uint64_t kog = (uint64_t)(uintptr_t)(ko + ((long long)bh * ST_ + s0) * DK_);
      const uint64_t vog = (uint64_t)(uintptr_t)(vo + ((long long)bh * ST_ + s0) * DK_);
      #pragma unroll
      for (int j = 0; j < 16; ++j) {
        uint32_t goff = (uint32_t)lane * 16u + (uint32_t)j * 512u;
        asm volatile("global_store_async_from_lds_b128 %0, %1, %2"
                     :: "v"(goff), "v"(klds + (uint32_t)j * 528u), "s"(kog)
                     : "memory");
        asm volatile("global_store_async_from_lds_b128 %0, %1, %2"
                     :: "v"(goff), "v"(vlds + (uint32_t)j * 528u), "s"(vog)
                     : "memory");
      }
    }

    // ---- online softmax (row r <-> matrix row r + hi*8) ----
    float p[8];
    #pragma unroll
    for (int r = 0; r < 8; ++r) {
      float v = sc[r] * scale;
      if (!wr && !sok) v = NEGINF;
      float mnew = fmaxf(m[r], rmax16(v));
      float a = __expf(m[r] - mnew);
      float pv_ = __expf(v - mnew);
      l[r] = l[r] * a + rsum16(pv_);
      m[r] = mnew;
      p[r] = pv_;
      #pragma unroll
      for (int nd = 0; nd < 8; ++nd) acc[nd][r] *= a;
    }

    // stash P in LDS (D-layout -> row-major), reload as A-fragments
    #pragma unroll
    for (int r = 0; r < 8; ++r) myp[(r + (hi << 3)) * 16 + ln] = p[r];

    // ---- ctx += P (A) x V_chunk (B) ----
    #pragma unroll
    for (int kk = 0; kk < 4; ++kk) {
      v2f pa = *reinterpret_cast<const v2f*>(myp + ln * 16 + (kk << 2) + koff);
      const int sr = (kk << 2) + koff;
      const bool ok0 = (s0 + sr) < ST_;
      const bool ok1 = (s0 + sr + 1) < ST_;
      #pragma unroll
      for (int nd = 0; nd < 8; ++nd) {
        int d = (nd << 4) + ln;
        v2f vb;
        if (wr) {
          vb.x = vbuf[sr * KVROW_ + d];
          vb.y = vbuf[(sr + 1) * KVROW_ + d];
        } else {
          vb.x = ok0 ? vp[sr * DK_ + d] : 0.f;
          vb.y = ok1 ? vp[(sr + 1) * DK_ + d] : 0.f;
        }
        acc[nd] = __builtin_amdgcn_wmma_f32_16x16x4_f32(false, pa, false, vb,
                                                        (short)0, acc[nd], false, false);
      }
    }
  }

  asm volatile("s_wait_asynccnt 0x0" ::: "memory");

  // ---- merge the 8 wave-partials (m, l, ctx) via LDS ----
  for (int i = threadIdx.x; i < 16 * DK_; i += 256) ctx_sh[i] = 0.f;
  if (threadIdx.x < 16) { m_sh[threadIdx.x] = NEGINF; l_sh[threadIdx.x] = 0.f; }
  __syncthreads();

  for (int w = 0; w < 8; ++w) {
    if (wave == w) {
      #pragma unroll
      for (int r = 0; r < 8; ++r) {
        int row = r + (hi << 3);
        float mg = m_sh[row];
        float mn = fmaxf(mg, m[r]);
        float ag = __expf(mg - mn);
        float aw = __expf(m[r] - mn);
        if (ln == 0) {
          m_sh[row] = mn;
          l_sh[row] = l_sh[row] * ag + l[r] * aw;
        }
        #pragma unroll
        for (int nd = 0; nd < 8; ++nd) {
          int col = (nd << 4) + ln;
          ctx_sh[row * DK_ + col] = ctx_sh[row * DK_ + col] * ag + acc[nd][r] * aw;
        }
      }
    }
    __syncthreads();
  }

  // normalize and emit ctx transposed to [B,T, H*dk] rows for the out-GEMM
  const int b = bh >> 4;
  const int h = bh & 15;
  for (int i = threadIdx.x; i < T_ * DK_; i += 256) {
    int t = i >> 7;
    int d = i & 127;
    ctxbuf[(b * T_ + t) * C_ + h * DK_ + d] = ctx_sh[t * DK_ + d] / l_sh[t];
  }
}

// ---------------------------------------------------------------------------
// Output projection: ctx[128,2048] @ w_out[2048,2048] + b_out -> out[128,2048]
// ---------------------------------------------------------------------------
__global__ __launch_bounds__(256)
void out_gemm_kernel(const float* __restrict__ ctx, const float* __restrict__ w,
                     const float* __restrict__ bias, float* __restrict__ out) {
  const int wave = threadIdx.x >> 5;
  const int lane = threadIdx.x & 31;
  const int tile = blockIdx.x * 8 + wave;          // 1024 tiles = 8 mt x 128 nt
  const int nt = tile % 128;
  const int mt = tile / 128;
  const int ln = lane & 15;
  const int koff = (lane >> 4) << 1;
  const int arow = mt * 16 + ln;
  const int bcol = nt * 16 + ln;

  v8f acc = {};
  for (int k = 0; k < C_; k += 4) {
    v2f a = *reinterpret_cast<const v2f*>(ctx + arow * C_ + k + koff);
    v2f b;
    b.x = w[(k + koff) * C_ + bcol];
    b.y = w[(k + koff + 1) * C_ + bcol];
    acc = __builtin_amdgcn_wmma_f32_16x16x4_f32(false, a, false, b, (short)0,
                                                acc, false, false);
  }
  const float bv = bias[bcol];
  const int rbase = mt * 16 + ((lane >> 4) << 3);
  #pragma unroll
  for (int r = 0; r < 8; ++r)
    out[(rbase + r) * C_ + bcol] = acc[r] + bv;
}

// ---------------------------------------------------------------------------
extern "C" void kernel_launch(void* const* d_in, const int* in_sizes, int n_in,
                              void* d_out, int out_size, void* d_ws, size_t ws_size,
                              hipStream_t stream) {
  (void)in_sizes; (void)n_in; (void)out_size; (void)ws_size;
  const float* x      = (const float*)d_in[0];
  const float* past_k = (const float*)d_in[1];
  const float* past_v = (const float*)d_in[2];
  const float* w_qkv  = (const float*)d_in[3];
  const float* b_qkv  = (const float*)d_in[4];
  const float* w_out  = (const float*)d_in[5];
  const float* b_out  = (const float*)d_in[6];

  float* out = (float*)d_out;                               // [16,8,2048]
  float* ko  = out + (long long)B_ * T_ * C_;               // [16,16,4104,128]
  float* vo  = ko + (long long)B_ * H_ * ST_ * DK_;

  float* qbuf   = (float*)d_ws;                             // 1 MB
  float* ctxbuf = qbuf + (long long)B_ * H_ * T_ * DK_;     // 1 MB

  qkv_gemm_kernel<<<384, 256, 0, stream>>>(x, w_qkv, b_qkv, qbuf, ko, vo);
  attn_kernel<<<B_ * H_, 256, SM_TOT_ * sizeof(float), stream>>>(
      qbuf, past_k, past_v, ko, vo, ctxbuf);
  out_gemm_kernel<<<128, 256, 0, stream>>>(ctxbuf, w_out, b_out, out);
}